// Autoencoder_44916767981863
// MI455X (gfx1250) — compile-verified
//
#include <hip/hip_runtime.h>
#include <hip/hip_bf16.h>

// ---------------- SLAYER constants (fp32) ----------------
#define THETA      10.0f
#define SCALE_REF  2.0f
#define D_SR       0.90483741803595952f   // exp(-1/10)
#define D_REF      0.36787944117144233f   // exp(-1)
#define PSP_SCALE  0.27182818284590452f   // e / tau_sr
#define POOL_SCALE 2.75f                  // 1.1 * theta / 4
#define TT 256                            // time steps (contiguous axis)

typedef float v2f __attribute__((ext_vector_type(2)));
typedef float v8f __attribute__((ext_vector_type(8)));

// =====================================================================
// WMMA conv: per-timestep 2D conv as GEMM.
//   M = 16 output channels, N = 16 consecutive timesteps,
//   K = Cin*KH*KW reduced in chunks of 4 via V_WMMA_F32_16X16X4_F32.
// One wave per (b, y, x, co-tile, t-tile).
// =====================================================================
template<int CIN, int COUT, int KH, int KW, int PAD, int H, int W>
__global__ void conv_wmma_kernel(const float* __restrict__ in,
                                 const float* __restrict__ wgt,
                                 float* __restrict__ out,
                                 int totalTiles)
{
    constexpr int R    = CIN * KH * KW;
    constexpr int nCoT = (COUT + 15) / 16;
    constexpr int nTt  = TT / 16;

    const int wave = threadIdx.x >> 5;
    const int lane = threadIdx.x & 31;
    int tile = blockIdx.x * (blockDim.x >> 5) + wave;
    if (tile >= totalTiles) return;           // whole-wave exit only

    int tmp = tile;
    const int tt  = tmp % nTt;  tmp /= nTt;
    const int xo  = tmp % W;    tmp /= W;
    const int yo  = tmp % H;    tmp /= H;
    const int coT = tmp % nCoT; tmp /= nCoT;
    const int bb  = tmp;

    const int t0  = tt * 16;
    const int co0 = coT * 16;
    const int hi  = lane >> 4;    // 0 or 1
    const int n   = lane & 15;    // N index (timestep) and A's M index

    v8f acc = {};

    for (int r0 = 0; r0 < R; r0 += 4) {
        v2f a, b;
        #pragma unroll
        for (int j = 0; j < 2; ++j) {
            const int rr = r0 + j + 2 * hi;   // K index held by this lane/VGPR
            float av = 0.0f, bv = 0.0f;
            if (rr < R) {
                const int ci  = rr / (KH * KW);
                const int rem = rr % (KH * KW);
                const int ky  = rem / KW;
                const int kx  = rem % KW;
                // A[m=n][k=rr] = w[co0+n, ci, ky, kx]  (zero-pad co >= COUT)
                if (co0 + n < COUT)
                    av = wgt[(((co0 + n) * CIN + ci) * KH + ky) * KW + kx];
                // B[k=rr][n] = in[bb, ci, yo+ky-PAD, xo+kx-PAD, t0+n]
                const int yy = yo + ky - PAD;
                const int xx = xo + kx - PAD;
                if (yy >= 0 && yy < H && xx >= 0 && xx < W)
                    bv = in[(((bb * CIN + ci) * H + yy) * W + xx) * TT + t0 + n];
            }
            a[j] = av;
            b[j] = bv;
        }
        acc = __builtin_amdgcn_wmma_f32_16x16x4_f32(
            /*neg_a=*/false, a, /*neg_b=*/false, b,
            /*c_mod=*/(short)0, acc, /*reuse_a=*/false, /*reuse_b=*/false);
    }

    #pragma unroll
    for (int v = 0; v < 8; ++v) {
        const int co = co0 + v + 8 * hi;      // C/D layout: VGPR v -> M = v + 8*hi
        if (co < COUT)
            out[(((bb * COUT + co) * H + yo) * W + xo) * TT + t0 + n] = acc[v];
    }
}

// =====================================================================
// Serial-in-time elementwise kernels: one thread per (b,c,h,w) row of
// 256 contiguous floats, streamed as float4.
// =====================================================================

// out = psp(x): y1 = d*y1 + x ; y2 = d*y2 + y1 ; out = (e/tau)*(y2-y1)
__global__ void psp_kernel(const float* __restrict__ in,
                           float* __restrict__ out, int rows)
{
    int idx = blockIdx.x * blockDim.x + threadIdx.x;
    if (idx >= rows) return;
    const float4* ip = (const float4*)(in + (size_t)idx * TT);
    float4* op = (float4*)(out + (size_t)idx * TT);
    float y1 = 0.f, y2 = 0.f;
    for (int i = 0; i < TT / 4; ++i) {
        float4 v = ip[i], o;
        y1 = D_SR * y1 + v.x; y2 = D_SR * y2 + y1; o.x = PSP_SCALE * (y2 - y1);
        y1 = D_SR * y1 + v.y; y2 = D_SR * y2 + y1; o.y = PSP_SCALE * (y2 - y1);
        y1 = D_SR * y1 + v.z; y2 = D_SR * y2 + y1; o.z = PSP_SCALE * (y2 - y1);
        y1 = D_SR * y1 + v.w; y2 = D_SR * y2 + y1; o.w = PSP_SCALE * (y2 - y1);
        op[i] = o;
    }
}

__device__ __forceinline__ float spike_psp_step(float u, float& r,
                                                float& y1, float& y2)
{
    float s = (u + r >= THETA) ? 1.0f : 0.0f;
    r = D_REF * (r - SCALE_REF * THETA * s);
    y1 = D_SR * y1 + s;
    y2 = D_SR * y2 + y1;
    return PSP_SCALE * (y2 - y1);
}

// out = psp(spike(u))
__global__ void spike_psp_kernel(const float* __restrict__ in,
                                 float* __restrict__ out, int rows)
{
    int idx = blockIdx.x * blockDim.x + threadIdx.x;
    if (idx >= rows) return;
    const float4* ip = (const float4*)(in + (size_t)idx * TT);
    float4* op = (float4*)(out + (size_t)idx * TT);
    float r = 0.f, y1 = 0.f, y2 = 0.f;
    for (int i = 0; i < TT / 4; ++i) {
        float4 v = ip[i], o;
        o.x = spike_psp_step(v.x, r, y1, y2);
        o.y = spike_psp_step(v.y, r, y1, y2);
        o.z = spike_psp_step(v.z, r, y1, y2);
        o.w = spike_psp_step(v.w, r, y1, y2);
        op[i] = o;
    }
}

// out = sumpool2(psp(spike(u))) : 4 fused chains per pooled output row
__global__ void spike_psp_pool_kernel(const float* __restrict__ in,
                                      float* __restrict__ out,
                                      int B, int C, int H, int W)
{
    const int HO = H / 2, WO = W / 2;
    const int rows = B * C * HO * WO;
    int idx = blockIdx.x * blockDim.x + threadIdx.x;
    if (idx >= rows) return;
    int tmp = idx;
    const int ox = tmp % WO; tmp /= WO;
    const int oy = tmp % HO; tmp /= HO;
    const int c  = tmp % C;  tmp /= C;
    const int b  = tmp;

    const float4* ip[4];
    #pragma unroll
    for (int k = 0; k < 4; ++k) {
        const int iy = 2 * oy + (k >> 1);
        const int ix = 2 * ox + (k & 1);
        ip[k] = (const float4*)(in + ((size_t)(((b * C + c) * H + iy) * W + ix)) * TT);
    }
    float4* op = (float4*)(out + (size_t)idx * TT);

    float r[4] = {0.f, 0.f, 0.f, 0.f};
    float y1[4] = {0.f, 0.f, 0.f, 0.f};
    float y2[4] = {0.f, 0.f, 0.f, 0.f};
    for (int i = 0; i < TT / 4; ++i) {
        float4 v0 = ip[0][i], v1 = ip[1][i], v2 = ip[2][i], v3 = ip[3][i];
        float4 o;
        o.x = POOL_SCALE * (spike_psp_step(v0.x, r[0], y1[0], y2[0]) +
                            spike_psp_step(v1.x, r[1], y1[1], y2[1]) +
                            spike_psp_step(v2.x, r[2], y1[2], y2[2]) +
                            spike_psp_step(v3.x, r[3], y1[3], y2[3]));
        o.y = POOL_SCALE * (spike_psp_step(v0.y, r[0], y1[0], y2[0]) +
                            spike_psp_step(v1.y, r[1], y1[1], y2[1]) +
                            spike_psp_step(v2.y, r[2], y1[2], y2[2]) +
                            spike_psp_step(v3.y, r[3], y1[3], y2[3]));
        o.z = POOL_SCALE * (spike_psp_step(v0.z, r[0], y1[0], y2[0]) +
                            spike_psp_step(v1.z, r[1], y1[1], y2[1]) +
                            spike_psp_step(v2.z, r[2], y1[2], y2[2]) +
                            spike_psp_step(v3.z, r[3], y1[3], y2[3]));
        o.w = POOL_SCALE * (spike_psp_step(v0.w, r[0], y1[0], y2[0]) +
                            spike_psp_step(v1.w, r[1], y1[1], y2[1]) +
                            spike_psp_step(v2.w, r[2], y1[2], y2[2]) +
                            spike_psp_step(v3.w, r[3], y1[3], y2[3]));
        op[i] = o;
    }
}

// out = upsample2_bilinear(psp(spike(u))) with JAX half-pixel convention:
// even out = 0.25*p[i-1] + 0.75*p[i]; odd out = 0.75*p[i] + 0.25*p[i+1]
// (indices clamped -> equals jax.image.resize edge renormalization)
__global__ void spike_psp_up_kernel(const float* __restrict__ in,
                                    float* __restrict__ out,
                                    int B, int C, int H, int W)
{
    const int HO = 2 * H, WO = 2 * W;
    const int rows = B * C * HO * WO;
    int idx = blockIdx.x * blockDim.x + threadIdx.x;
    if (idx >= rows) return;
    int tmp = idx;
    const int ox = tmp % WO; tmp /= WO;
    const int oy = tmp % HO; tmp /= HO;
    const int c  = tmp % C;  tmp /= C;
    const int b  = tmp;

    int iy0, iy1, ix0, ix1;
    float wy0, wy1, wx0, wx1;
    if (oy & 1) { iy0 = oy >> 1; iy1 = (oy >> 1) + 1; wy0 = 0.75f; wy1 = 0.25f; }
    else        { iy0 = (oy >> 1) - 1; iy1 = oy >> 1; wy0 = 0.25f; wy1 = 0.75f; }
    if (ox & 1) { ix0 = ox >> 1; ix1 = (ox >> 1) + 1; wx0 = 0.75f; wx1 = 0.25f; }
    else        { ix0 = (ox >> 1) - 1; ix1 = ox >> 1; wx0 = 0.25f; wx1 = 0.75f; }
    iy0 = iy0 < 0 ? 0 : (iy0 > H - 1 ? H - 1 : iy0);
    iy1 = iy1 < 0 ? 0 : (iy1 > H - 1 ? H - 1 : iy1);
    ix0 = ix0 < 0 ? 0 : (ix0 > W - 1 ? W - 1 : ix0);
    ix1 = ix1 < 0 ? 0 : (ix1 > W - 1 ? W - 1 : ix1);

    const float w00 = wy0 * wx0, w01 = wy0 * wx1, w10 = wy1 * wx0, w11 = wy1 * wx1;
    const float4* ip[4];
    ip[0] = (const float4*)(in + ((size_t)(((b * C + c) * H + iy0) * W + ix0)) * TT);
    ip[1] = (const float4*)(in + ((size_t)(((b * C + c) * H + iy0) * W + ix1)) * TT);
    ip[2] = (const float4*)(in + ((size_t)(((b * C + c) * H + iy1) * W + ix0)) * TT);
    ip[3] = (const float4*)(in + ((size_t)(((b * C + c) * H + iy1) * W + ix1)) * TT);
    float4* op = (float4*)(out + (size_t)idx * TT);

    float r[4] = {0.f, 0.f, 0.f, 0.f};
    float y1[4] = {0.f, 0.f, 0.f, 0.f};
    float y2[4] = {0.f, 0.f, 0.f, 0.f};
    for (int i = 0; i < TT / 4; ++i) {
        float4 v0 = ip[0][i], v1 = ip[1][i], v2 = ip[2][i], v3 = ip[3][i];
        float4 o;
        o.x = w00 * spike_psp_step(v0.x, r[0], y1[0], y2[0]) +
              w01 * spike_psp_step(v1.x, r[1], y1[1], y2[1]) +
              w10 * spike_psp_step(v2.x, r[2], y1[2], y2[2]) +
              w11 * spike_psp_step(v3.x, r[3], y1[3], y2[3]);
        o.y = w00 * spike_psp_step(v0.y, r[0], y1[0], y2[0]) +
              w01 * spike_psp_step(v1.y, r[1], y1[1], y2[1]) +
              w10 * spike_psp_step(v2.y, r[2], y1[2], y2[2]) +
              w11 * spike_psp_step(v3.y, r[3], y1[3], y2[3]);
        o.z = w00 * spike_psp_step(v0.z, r[0], y1[0], y2[0]) +
              w01 * spike_psp_step(v1.z, r[1], y1[1], y2[1]) +
              w10 * spike_psp_step(v2.z, r[2], y1[2], y2[2]) +
              w11 * spike_psp_step(v3.z, r[3], y1[3], y2[3]);
        o.w = w00 * spike_psp_step(v0.w, r[0], y1[0], y2[0]) +
              w01 * spike_psp_step(v1.w, r[1], y1[1], y2[1]) +
              w10 * spike_psp_step(v2.w, r[2], y1[2], y2[2]) +
              w11 * spike_psp_step(v3.w, r[3], y1[3], y2[3]);
        op[i] = o;
    }
}

// final: out = spike(u) only
__global__ void spike_final_kernel(const float* __restrict__ in,
                                   float* __restrict__ out, int rows)
{
    int idx = blockIdx.x * blockDim.x + threadIdx.x;
    if (idx >= rows) return;
    const float4* ip = (const float4*)(in + (size_t)idx * TT);
    float4* op = (float4*)(out + (size_t)idx * TT);
    float r = 0.f;
    for (int i = 0; i < TT / 4; ++i) {
        float4 v = ip[i], o;
        o.x = (v.x + r >= THETA) ? 1.0f : 0.0f; r = D_REF * (r - SCALE_REF * THETA * o.x);
        o.y = (v.y + r >= THETA) ? 1.0f : 0.0f; r = D_REF * (r - SCALE_REF * THETA * o.y);
        o.z = (v.z + r >= THETA) ? 1.0f : 0.0f; r = D_REF * (r - SCALE_REF * THETA * o.z);
        o.w = (v.w + r >= THETA) ? 1.0f : 0.0f; r = D_REF * (r - SCALE_REF * THETA * o.w);
        op[i] = o;
    }
}

// =====================================================================
extern "C" void kernel_launch(void* const* d_in, const int* in_sizes, int n_in,
                              void* d_out, int out_size, void* d_ws, size_t ws_size,
                              hipStream_t stream)
{
    const float* x     = (const float*)d_in[0]; // [4,1,32,32,256]
    const float* w1    = (const float*)d_in[1]; // [16,1,5,5,1]
    const float* w2    = (const float*)d_in[2]; // [32,16,3,3,1]
    const float* w3    = (const float*)d_in[3]; // [64,32,3,3,1]
    const float* w4    = (const float*)d_in[4]; // [32,64,3,3,1]
    const float* w_out = (const float*)d_in[5]; // [1,32,1,1,1]
    float* out = (float*)d_out;

    // ping-pong buffers, 33.55M floats (134 MB) each; whole chain fits in L2
    float* A = (float*)d_ws;
    float* B = A + (size_t)33554432;

    const dim3 blk(256);
    auto rg = [](int rows) { return dim3((unsigned)((rows + 255) / 256)); };
    auto cg = [](int tiles) { return dim3((unsigned)((tiles + 7) / 8)); }; // 8 waves/block

    // a1 = psp(input)                               [4,1,32,32,256]
    psp_kernel<<<rg(4096), blk, 0, stream>>>(x, A, 4096);
    // c1 = conv(a1, w1, pad=2)                      [4,16,32,32,256]
    { int tiles = 4 * 1 * 32 * 32 * 16;
      conv_wmma_kernel<1, 16, 5, 5, 2, 32, 32><<<cg(tiles), blk, 0, stream>>>(A, w1, B, tiles); }
    // q2 = pool(psp(spike(c1)))                     [4,16,16,16,256]
    spike_psp_pool_kernel<<<rg(16384), blk, 0, stream>>>(B, A, 4, 16, 32, 32);
    // a3 = psp(spike(q2))
    spike_psp_kernel<<<rg(16384), blk, 0, stream>>>(A, B, 16384);
    // c2 = conv(a3, w2, pad=1)                      [4,32,16,16,256]
    { int tiles = 4 * 2 * 16 * 16 * 16;
      conv_wmma_kernel<16, 32, 3, 3, 1, 16, 16><<<cg(tiles), blk, 0, stream>>>(B, w2, A, tiles); }
    // q4 = pool(psp(spike(c2)))                     [4,32,8,8,256]
    spike_psp_pool_kernel<<<rg(8192), blk, 0, stream>>>(A, B, 4, 32, 16, 16);
    // a5 = psp(spike(q4))
    spike_psp_kernel<<<rg(8192), blk, 0, stream>>>(B, A, 8192);
    // c3 = conv(a5, w3, pad=1)                      [4,64,8,8,256]
    { int tiles = 4 * 4 * 8 * 8 * 16;
      conv_wmma_kernel<32, 64, 3, 3, 1, 8, 8><<<cg(tiles), blk, 0, stream>>>(A, w3, B, tiles); }
    // q6 = upsample(psp(spike(c3)))                 [4,64,16,16,256]
    spike_psp_up_kernel<<<rg(65536), blk, 0, stream>>>(B, A, 4, 64, 8, 8);
    // a7 = psp(spike(q6))
    spike_psp_kernel<<<rg(65536), blk, 0, stream>>>(A, B, 65536);
    // c4 = conv(a7, w4, pad=1)                      [4,32,16,16,256]
    { int tiles = 4 * 2 * 16 * 16 * 16;
      conv_wmma_kernel<64, 32, 3, 3, 1, 16, 16><<<cg(tiles), blk, 0, stream>>>(B, w4, A, tiles); }
    // q8 = upsample(psp(spike(c4)))                 [4,32,32,32,256]
    spike_psp_up_kernel<<<rg(131072), blk, 0, stream>>>(A, B, 4, 32, 16, 16);
    // a9 = psp(spike(q8))
    spike_psp_kernel<<<rg(131072), blk, 0, stream>>>(B, A, 131072);
    // c5 = conv(a9, w_out, pad=0)                   [4,1,32,32,256]
    { int tiles = 4 * 1 * 32 * 32 * 16;
      conv_wmma_kernel<32, 1, 1, 1, 0, 32, 32><<<cg(tiles), blk, 0, stream>>>(A, w_out, B, tiles); }
    // out = spike(c5)
    spike_final_kernel<<<rg(4096), blk, 0, stream>>>(B, out, 4096);
}